// Slice_windows_42030549958970
// MI455X (gfx1250) — compile-verified
//
#include <hip/hip_runtime.h>

// ---------------------------------------------------------------------------
// Slice_windows entropy pyramid for x:(32,1,1024,1024) f32 -> out:(32,8) f32
// Single HBM pass, hierarchical pooling; level-1 (2x2) pooling done with
// V_WMMA_F32_16X16X4_F32 (D = A x B_pool + C chained over K), rest in
// registers/LDS. MI455X (gfx1250), wave32.
// ---------------------------------------------------------------------------

typedef __attribute__((ext_vector_type(2))) float v2f;
typedef __attribute__((ext_vector_type(8))) float v8f;

#define LDSW 132   // padded LDS row stride (floats): conflict-free b64 A loads

__device__ __forceinline__ float winH(float s, float invn) {
  // Shannon entropy of Bernoulli(p1 = s/n), plogp(0) := 0, base-2 log.
  float p1 = s * invn;
  float p0 = 1.0f - p1;
  float h = 0.0f;
  if (p1 > 0.0f) h -= p1 * __log2f(p1);
  if (p0 > 0.0f) h -= p0 * __log2f(p0);
  return h;
}

__global__ __launch_bounds__(256) void entropy_stage1(
    const float* __restrict__ x, float* __restrict__ part)
{
  const int bx   = blockIdx.x;
  const int img  = bx >> 6;        // 32 images
  const int tile = bx & 63;        // 8x8 tiles of 128x128
  const int tr   = (tile >> 3) * 128;
  const int tc   = (tile & 7) * 128;

  __shared__ float tb[64 * LDSW];  // 64-row half-tile staging (~33.8 KB)
  __shared__ float ug[16 * 16];    // 8x8-window sums over the 128x128 tile
  __shared__ float g16[64];
  __shared__ float g32[16];
  __shared__ float g64[4];
  __shared__ float red[7 * 256];   // per-level block reduction

  const int t    = threadIdx.x;
  const int lane = t & 31;
  const int wave = t >> 5;

  float e0 = 0.f, e1 = 0.f, e2 = 0.f, e3 = 0.f, e4 = 0.f, e5 = 0.f, e6 = 0.f;

  const float* src = x + (size_t)img * (1024u * 1024u) + (size_t)tr * 1024 + tc;

  for (int half = 0; half < 2; ++half) {
    // ---- stage 64 rows x 128 cols into LDS (coalesced b128) ----
    const float* s0 = src + (size_t)(half * 64) * 1024;
    #pragma unroll
    for (int i = 0; i < 8; ++i) {
      int lin = i * 256 + t;          // 0..2047
      int r   = lin >> 5;             // 0..63
      int c4  = lin & 31;             // 0..31 float4 columns
      float4 v = *(const float4*)(s0 + (size_t)r * 1024 + c4 * 4);
      *(float4*)&tb[r * LDSW + c4 * 4] = v;
    }
    __syncthreads();

    // wave task: 16 rows x 64 cols of the half-tile
    const int rstrip = (wave & 3) * 16;
    const int chalf  = (wave >> 2) * 64;
    const int arow   = rstrip + (lane & 15);

    for (int region = 0; region < 2; ++region) {
      const int colbase = chalf + region * 32;

      // --- WMMA horizontal-pair pooling of a 16x32 region ---
      // D[m][p] = x[m][2p] + x[m][2p+1], via 8 chained 16x16x4 f32 WMMAs
      // with one-hot pooling B (B_j[k][n] = 1 iff n == 2j + k/2).
      v8f d = {0.f, 0.f, 0.f, 0.f, 0.f, 0.f, 0.f, 0.f};
      #pragma unroll
      for (int j = 0; j < 8; ++j) {
        // A 16x4 f32 layout: lane l -> (M=l%16, K=2*(l/16)+{0,1})
        v2f a = *(const v2f*)&tb[arow * LDSW + colbase + j * 4 + 2 * (lane >> 4)];
        float bb = (lane == 2 * j || lane == 2 * j + 17) ? 1.0f : 0.0f;
        v2f b = {bb, bb};
        d = __builtin_amdgcn_wmma_f32_16x16x4_f32(
                false, a, false, b, (short)0, d, false, false);
      }

      // level 0 (k=2, n=4): vertical pair of D rows (adjacent D VGPRs)
      float vq0 = d[0] + d[1], vq1 = d[2] + d[3];
      float vq2 = d[4] + d[5], vq3 = d[6] + d[7];
      const float i4 = 0.25f;
      e0 += winH(vq0, i4) + winH(vq1, i4) + winH(vq2, i4) + winH(vq3, i4);

      // level 1 (k=4, n=16): 2x2 pool of the 2x2 sums (xor-1 butterfly)
      float t0 = vq0 + vq1, t1 = vq2 + vq3;
      float w0 = t0 + __shfl_xor(t0, 1, 32);
      float w1 = t1 + __shfl_xor(t1, 1, 32);
      if (!(lane & 1)) e1 += winH(w0, 1.f / 16.f) + winH(w1, 1.f / 16.f);

      // level 2 (k=8, n=64): xor-2 butterfly
      float s8 = w0 + w1;
      float u  = s8 + __shfl_xor(s8, 2, 32);
      if (!(lane & 3)) {
        e2 += winH(u, 1.f / 64.f);
        int urow = half * 8 + (wave & 3) * 2 + (lane >> 4);
        int ucol = (wave >> 2) * 8 + region * 4 + ((lane & 15) >> 2);
        ug[urow * 16 + ucol] = u;
      }
    }
    __syncthreads();
  }

  // level 3 (k=16, n=256): 2x2 pool of 16x16 grid of 8x8-sums
  if (t < 64) {
    int r = t >> 3, c = t & 7;
    float s = ug[(2 * r) * 16 + 2 * c]     + ug[(2 * r) * 16 + 2 * c + 1] +
              ug[(2 * r + 1) * 16 + 2 * c] + ug[(2 * r + 1) * 16 + 2 * c + 1];
    e3 += winH(s, 1.f / 256.f);
    g16[t] = s;
  }
  __syncthreads();
  // level 4 (k=32, n=1024)
  if (t < 16) {
    int r = t >> 2, c = t & 3;
    float s = g16[(2 * r) * 8 + 2 * c]     + g16[(2 * r) * 8 + 2 * c + 1] +
              g16[(2 * r + 1) * 8 + 2 * c] + g16[(2 * r + 1) * 8 + 2 * c + 1];
    e4 += winH(s, 1.f / 1024.f);
    g32[t] = s;
  }
  __syncthreads();
  // level 5 (k=64, n=4096)
  if (t < 4) {
    int r = t >> 1, c = t & 1;
    float s = g32[(2 * r) * 4 + 2 * c]     + g32[(2 * r) * 4 + 2 * c + 1] +
              g32[(2 * r + 1) * 4 + 2 * c] + g32[(2 * r + 1) * 4 + 2 * c + 1];
    e5 += winH(s, 1.f / 4096.f);
    g64[t] = s;
  }
  __syncthreads();
  // level 6 (k=128, n=16384): one window == the whole tile
  float tileSum = 0.f;
  if (t == 0) {
    tileSum = g64[0] + g64[1] + g64[2] + g64[3];
    e6 += winH(tileSum, 1.f / 16384.f);
  }

  // deterministic block tree-reduction of the 7 entropy partials
  red[0 * 256 + t] = e0; red[1 * 256 + t] = e1; red[2 * 256 + t] = e2;
  red[3 * 256 + t] = e3; red[4 * 256 + t] = e4; red[5 * 256 + t] = e5;
  red[6 * 256 + t] = e6;
  __syncthreads();
  for (int off = 128; off > 0; off >>= 1) {
    if (t < off) {
      #pragma unroll
      for (int l = 0; l < 7; ++l) red[l * 256 + t] += red[l * 256 + t + off];
    }
    __syncthreads();
  }

  float* slot = part + (size_t)(img * 64 + tile) * 8;
  if (t < 7)  slot[t] = red[t * 256];
  if (t == 0) slot[7] = tileSum;
}

__global__ __launch_bounds__(64) void entropy_stage2(
    const float* __restrict__ part, float* __restrict__ out)
{
  const int b = blockIdx.x;   // image
  const int t = threadIdx.x;  // tile (0..63)

  __shared__ float ts[64];
  __shared__ float red2[7 * 64];
  __shared__ float e8s[16];

  const float* p = part + (size_t)(b * 64 + t) * 8;
  #pragma unroll
  for (int l = 0; l < 7; ++l) red2[l * 64 + t] = p[l];
  ts[t] = p[7];
  __syncthreads();

  for (int off = 32; off > 0; off >>= 1) {
    if (t < off) {
      #pragma unroll
      for (int l = 0; l < 7; ++l) red2[l * 64 + t] += red2[l * 64 + t + off];
    }
    __syncthreads();
  }

  // level 7 (k=256, n=65536): 2x2 pool of the 8x8 grid of 128-tile sums
  if (t < 16) {
    int r = t >> 2, c = t & 3;
    float s = ts[(2 * r) * 8 + 2 * c]     + ts[(2 * r) * 8 + 2 * c + 1] +
              ts[(2 * r + 1) * 8 + 2 * c] + ts[(2 * r + 1) * 8 + 2 * c + 1];
    e8s[t] = winH(s, 1.f / 65536.f);
  }
  __syncthreads();

  if (t == 0) {
    #pragma unroll
    for (int l = 0; l < 7; ++l) {
      int k   = 2 << l;
      int hk  = 1024 / k;
      float cnt = (float)hk * (float)hk;
      out[b * 8 + l] = red2[l * 64] / cnt;
    }
    float e8 = 0.f;
    #pragma unroll
    for (int i = 0; i < 16; ++i) e8 += e8s[i];
    out[b * 8 + 7] = e8 * (1.f / 16.f);
  }
}

extern "C" void kernel_launch(void* const* d_in, const int* in_sizes, int n_in,
                              void* d_out, int out_size, void* d_ws, size_t ws_size,
                              hipStream_t stream) {
  (void)in_sizes; (void)n_in; (void)out_size; (void)ws_size;
  const float* x = (const float*)d_in[0];
  float* out  = (float*)d_out;
  float* part = (float*)d_ws;   // 32*64*8 floats = 64 KB scratch, fully rewritten

  entropy_stage1<<<dim3(32 * 64), dim3(256), 0, stream>>>(x, part);
  entropy_stage2<<<dim3(32), dim3(64), 0, stream>>>(part, out);
}